// E2E_53377853555300
// MI455X (gfx1250) — compile-verified
//
#include <hip/hip_runtime.h>

#define N_NODES 30000
#define N_EDGES 480000
#define EPSV 1e-5f

typedef __attribute__((ext_vector_type(2))) float v2f;
typedef __attribute__((ext_vector_type(8))) float v8f;

__device__ __forceinline__ v8f wmma_f32(v2f a, v2f b, v8f c) {
    // D = A(16x4) * B(4x16) + C, full fp32
    return __builtin_amdgcn_wmma_f32_16x16x4_f32(false, a, false, b, (short)0, c, false, false);
}

// C[M,Ncols] = opt_relu( A1[M,K1]@W1[K1,Ncols] + A2[M,K2]@W2[K2,Ncols] + bias )
// One wave -> 16x32 output tile. Grid must cover tiles exactly (EXEC all-1s for WMMA).
__global__ __launch_bounds__(128) void gemm_dual_wmma(
    const float* __restrict__ A1, const float* __restrict__ W1, int K1,
    const float* __restrict__ A2, const float* __restrict__ W2, int K2,
    const float* __restrict__ bias, float* __restrict__ C,
    int Ncols, int ntilesN, int do_relu)
{
    int wave = (blockIdx.x * blockDim.x + threadIdx.x) >> 5;
    int lane = threadIdx.x & 31;
    int mtile = wave / ntilesN;
    int ntile = wave - mtile * ntilesN;
    int m0 = mtile * 16;
    int n0 = ntile * 32;
    int lm = lane & 15;
    int lh = lane >> 4;            // selects K pair {0,1} vs {2,3}
    v8f acc0 = {}; v8f acc1 = {};

    for (int pass = 0; pass < 2; ++pass) {
        const float* A = pass ? A2 : A1;
        const float* W = pass ? W2 : W1;
        int K = pass ? K2 : K1;
        if (A == nullptr) continue;              // uniform branch, EXEC unchanged
        const float* arow  = A + (size_t)(m0 + lm) * K + 2 * lh;
        const float* wbase = W + (size_t)(2 * lh) * Ncols + n0 + lm;
        for (int k = 0; k < K; k += 4) {
            v2f a = *(const v2f*)(arow + k);     // A[m][k+2lh], A[m][k+2lh+1]
            const float* wp = wbase + (size_t)k * Ncols;
            v2f b0, b1;
            b0.x = wp[0];  b0.y = wp[Ncols];     // B[k+2lh][n], B[k+2lh+1][n]
            b1.x = wp[16]; b1.y = wp[Ncols + 16];
            acc0 = wmma_f32(a, b0, acc0);
            acc1 = wmma_f32(a, b1, acc1);
        }
    }
    int col0 = n0 + lm;
    float bb0 = bias ? bias[col0]      : 0.0f;
    float bb1 = bias ? bias[col0 + 16] : 0.0f;
    #pragma unroll
    for (int j = 0; j < 8; ++j) {
        int row = m0 + j + 8 * lh;               // C/D layout: vgpr j -> M=j (+8 for hi half)
        float v0 = acc0[j] + bb0;
        float v1 = acc1[j] + bb1;
        if (do_relu) { v0 = fmaxf(v0, 0.0f); v1 = fmaxf(v1, 0.0f); }
        C[(size_t)row * Ncols + col0]      = v0;
        C[(size_t)row * Ncols + col0 + 16] = v1;
    }
}

__global__ void zero_kernel(float* __restrict__ p, size_t n) {
    size_t i = (size_t)blockIdx.x * blockDim.x + threadIdx.x;
    if (i < n) p[i] = 0.0f;
}

// segment-max of relu'd values: values >= 0 so u32 bit pattern is order-preserving.
// Zero-init of agg implements where(deg>0, segmax, 0) exactly. One wave per edge.
__global__ __launch_bounds__(256) void scatter_max_kernel(
    const float* __restrict__ m, const int* __restrict__ src, const int* __restrict__ dst,
    unsigned int* __restrict__ agg, int C)
{
    int e    = (blockIdx.x * blockDim.x + threadIdx.x) >> 5;
    int lane = threadIdx.x & 31;
    int s = src[e], d = dst[e];
    const float* ms = m + (size_t)s * C;
    unsigned int* av = agg + (size_t)d * C;
    int vpl = C >> 5;
    for (int i = 0; i < vpl; ++i) {
        int c = lane + 32 * i;
        atomicMax(&av[c], __float_as_uint(ms[c]));
    }
}

// node head: y = LN(h1 @ npW + b); out node_pred = y, cls_sm = softmax(y). One wave/node.
__global__ __launch_bounds__(256) void node_head_kernel(
    const float* __restrict__ h1, const float* __restrict__ W, const float* __restrict__ b,
    const float* __restrict__ g, const float* __restrict__ beta,
    float* __restrict__ node_pred, float* __restrict__ cls_sm)
{
    int v    = (blockIdx.x * blockDim.x + threadIdx.x) >> 5;
    int lane = threadIdx.x & 31;
    const float* hp = h1 + (size_t)v * 256 + lane * 8;
    float p[5] = {0, 0, 0, 0, 0};
    #pragma unroll
    for (int j = 0; j < 8; ++j) {
        float x = hp[j];
        int kk = lane * 8 + j;
        #pragma unroll
        for (int c = 0; c < 5; ++c) p[c] += x * W[kk * 5 + c];
    }
    for (int off = 16; off > 0; off >>= 1)
        #pragma unroll
        for (int c = 0; c < 5; ++c) p[c] += __shfl_xor(p[c], off, 32);
    #pragma unroll
    for (int c = 0; c < 5; ++c) p[c] += b[c];
    float mu = (p[0] + p[1] + p[2] + p[3] + p[4]) * 0.2f;
    float var = 0.0f;
    #pragma unroll
    for (int c = 0; c < 5; ++c) { float dl = p[c] - mu; var += dl * dl; }
    float rstd = rsqrtf(var * 0.2f + EPSV);
    float y[5];
    #pragma unroll
    for (int c = 0; c < 5; ++c) y[c] = g[c] * (p[c] - mu) * rstd + beta[c];
    float mx = y[0];
    #pragma unroll
    for (int c = 1; c < 5; ++c) mx = fmaxf(mx, y[c]);
    float ex[5], sume = 0.0f;
    #pragma unroll
    for (int c = 0; c < 5; ++c) { ex[c] = __expf(y[c] - mx); sume += ex[c]; }
    float inv = 1.0f / sume;
    if (lane == 0) {
        #pragma unroll
        for (int c = 0; c < 5; ++c) {
            node_pred[(size_t)v * 5 + c] = y[c];
            cls_sm[(size_t)v * 5 + c]    = ex[c] * inv;
        }
    }
}

// A[v] += cls[v]@Wb + b1 ; B[v] += cls[v]@We. One wave per node.
__global__ __launch_bounds__(256) void node_proj_epilogue(
    float* __restrict__ A, float* __restrict__ B, const float* __restrict__ cls,
    const float* __restrict__ Wb, const float* __restrict__ We, const float* __restrict__ b1)
{
    int v    = (blockIdx.x * blockDim.x + threadIdx.x) >> 5;
    int lane = threadIdx.x & 31;
    float c0 = cls[(size_t)v * 5 + 0], c1 = cls[(size_t)v * 5 + 1], c2 = cls[(size_t)v * 5 + 2];
    float c3 = cls[(size_t)v * 5 + 3], c4 = cls[(size_t)v * 5 + 4];
    #pragma unroll
    for (int i = 0; i < 8; ++i) {
        int col = lane + 32 * i;
        size_t idx = (size_t)v * 256 + col;
        float a = A[idx] + b1[col];
        a += c0 * Wb[col] + c1 * Wb[256 + col] + c2 * Wb[512 + col] + c3 * Wb[768 + col] + c4 * Wb[1024 + col];
        A[idx] = a;
        float bv = B[idx];
        bv += c0 * We[col] + c1 * We[256 + col] + c2 * We[512 + col] + c3 * We[768 + col] + c4 * We[1024 + col];
        B[idx] = bv;
    }
}

// Per-edge: x = A[src] + B[dst] + efeat@Wc; relu(LN(x)); out = x@W2 + b2. One wave per edge.
__global__ __launch_bounds__(256) void edge_kernel(
    const float* __restrict__ A, const float* __restrict__ B,
    const float* __restrict__ efeat, const int* __restrict__ src, const int* __restrict__ dst,
    const float* __restrict__ Wc, const float* __restrict__ g, const float* __restrict__ beta,
    const float* __restrict__ W2, const float* __restrict__ b2, float* __restrict__ out)
{
    int e    = (blockIdx.x * blockDim.x + threadIdx.x) >> 5;
    int lane = threadIdx.x & 31;
    int s = src[e], d = dst[e];
    float ef[6];
    #pragma unroll
    for (int i = 0; i < 6; ++i) ef[i] = efeat[(size_t)e * 6 + i];
    const float* ap = A + (size_t)s * 256 + lane * 8;
    const float* bp = B + (size_t)d * 256 + lane * 8;
    float x[8];
    float ssum = 0.0f, ssq = 0.0f;
    #pragma unroll
    for (int j = 0; j < 8; ++j) {
        int col = lane * 8 + j;
        float xx = ap[j] + bp[j];
        #pragma unroll
        for (int i = 0; i < 6; ++i) xx += ef[i] * Wc[i * 256 + col];
        x[j] = xx; ssum += xx; ssq += xx * xx;
    }
    for (int off = 16; off > 0; off >>= 1) {
        ssum += __shfl_xor(ssum, off, 32);
        ssq  += __shfl_xor(ssq,  off, 32);
    }
    float mu = ssum * (1.0f / 256.0f);
    float var = ssq * (1.0f / 256.0f) - mu * mu;
    float rstd = rsqrtf(var + EPSV);
    float y0 = 0.0f, y1 = 0.0f;
    #pragma unroll
    for (int j = 0; j < 8; ++j) {
        int col = lane * 8 + j;
        float xn = fmaxf(g[col] * (x[j] - mu) * rstd + beta[col], 0.0f);
        y0 += xn * W2[col * 2 + 0];
        y1 += xn * W2[col * 2 + 1];
    }
    for (int off = 16; off > 0; off >>= 1) {
        y0 += __shfl_xor(y0, off, 32);
        y1 += __shfl_xor(y1, off, 32);
    }
    if (lane == 0) {
        out[(size_t)e * 2 + 0] = y0 + b2[0];
        out[(size_t)e * 2 + 1] = y1 + b2[1];
    }
}

extern "C" void kernel_launch(void* const* d_in, const int* in_sizes, int n_in,
                              void* d_out, int out_size, void* d_ws, size_t ws_size,
                              hipStream_t stream)
{
    (void)in_sizes; (void)n_in; (void)out_size; (void)ws_size;
    const float* h      = (const float*)d_in[0];
    const float* efeat  = (const float*)d_in[1];
    const int*   src    = (const int*)d_in[2];
    const int*   dst    = (const int*)d_in[3];
    const float* encWp  = (const float*)d_in[4];
    const float* encbp  = (const float*)d_in[5];
    const float* encWs  = (const float*)d_in[6];
    const float* encbs  = (const float*)d_in[7];
    const float* encWn  = (const float*)d_in[8];
    const float* npW    = (const float*)d_in[9];
    const float* npb    = (const float*)d_in[10];
    const float* npg    = (const float*)d_in[11];
    const float* npbeta = (const float*)d_in[12];
    const float* epW1   = (const float*)d_in[13];
    const float* epb1   = (const float*)d_in[14];
    const float* epg    = (const float*)d_in[15];
    const float* epbeta = (const float*)d_in[16];
    const float* epW2   = (const float*)d_in[17];
    const float* epb2   = (const float*)d_in[18];
    const float* decWp  = (const float*)d_in[19];
    const float* decbp  = (const float*)d_in[20];
    const float* decWs  = (const float*)d_in[21];
    const float* decbs  = (const float*)d_in[22];
    const float* decWn  = (const float*)d_in[23];

    float* ws = (float*)d_ws;
    float* X0 = ws;                              // N*256: m -> A -> m2
    float* X1 = X0 + (size_t)N_NODES * 256;      // N*256: agg -> B -> agg2
    float* X2 = X1 + (size_t)N_NODES * 256;      // N*256: h1
    float* X3 = X2 + (size_t)N_NODES * 256;      // N*5  : cls_sm

    float* out_node = (float*)d_out;                   // [N,5]
    float* out_edge = out_node + (size_t)N_NODES * 5;  // [E,2]
    float* out_h2   = out_edge + (size_t)N_EDGES * 2;  // [N,128]

    // ---- encoder ----
    // m = relu(h @ encWp + encbp)   [N,128]; 7500 waves
    gemm_dual_wmma<<<1875, 128, 0, stream>>>(h, encWp, 128, nullptr, nullptr, 0,
                                             encbp, X0, 128, 4, 1);
    zero_kernel<<<(int)(((size_t)N_NODES * 128 + 255) / 256), 256, 0, stream>>>(X1, (size_t)N_NODES * 128);
    scatter_max_kernel<<<N_EDGES / 8, 256, 0, stream>>>(X0, src, dst, (unsigned int*)X1, 128);
    // h1 = relu(h @ encWs + agg @ encWn + encbs)  [N,256]; 15000 waves
    gemm_dual_wmma<<<3750, 128, 0, stream>>>(h, encWs, 128, X1, encWn, 128,
                                             encbs, X2, 256, 8, 1);
    // ---- node head ----
    node_head_kernel<<<N_NODES / 8, 256, 0, stream>>>(X2, npW, npb, npg, npbeta, out_node, X3);
    // ---- edge head (decomposed concat GEMM) ----
    gemm_dual_wmma<<<3750, 128, 0, stream>>>(X2, epW1, 256, nullptr, nullptr, 0,
                                             nullptr, X0, 256, 8, 0);                 // A = h1@W1[0:256]
    gemm_dual_wmma<<<3750, 128, 0, stream>>>(X2, epW1 + (size_t)267 * 256, 256, nullptr, nullptr, 0,
                                             nullptr, X1, 256, 8, 0);                 // B = h1@W1[267:523]
    node_proj_epilogue<<<N_NODES / 8, 256, 0, stream>>>(X0, X1, X3,
                                                        epW1 + (size_t)256 * 256,
                                                        epW1 + (size_t)523 * 256, epb1);
    edge_kernel<<<N_EDGES / 8, 256, 0, stream>>>(X0, X1, efeat, src, dst,
                                                 epW1 + (size_t)261 * 256,
                                                 epg, epbeta, epW2, epb2, out_edge);
    // ---- decoder ----
    gemm_dual_wmma<<<3750, 128, 0, stream>>>(X2, decWp, 256, nullptr, nullptr, 0,
                                             decbp, X0, 256, 8, 1);                   // m2
    zero_kernel<<<(int)(((size_t)N_NODES * 256 + 255) / 256), 256, 0, stream>>>(X1, (size_t)N_NODES * 256);
    scatter_max_kernel<<<N_EDGES / 8, 256, 0, stream>>>(X0, src, dst, (unsigned int*)X1, 256);
    // h2 = relu(h1 @ decWs + agg2 @ decWn + decbs)  [N,128]; 7500 waves
    gemm_dual_wmma<<<1875, 128, 0, stream>>>(X2, decWs, 256, X1, decWn, 256,
                                             decbs, out_h2, 128, 4, 1);
}